// GaussianState_47631187313223
// MI455X (gfx1250) — compile-verified
//
#include <hip/hip_runtime.h>
#include <math.h>

#define NSPIN 16
#define NMAJ  32
#define NPF   64
#define BATCH 256

typedef __attribute__((ext_vector_type(2))) float v2f;
typedef __attribute__((ext_vector_type(8))) float v8f;
typedef __attribute__((ext_vector_type(4))) int   i4v;

__device__ __forceinline__ v8f wmma_f32_k4(v8f c, v2f a, v2f b) {
  // D = A(16x4) * B(4x16) + C  ; f32 matrix core op, wave32
  return __builtin_amdgcn_wmma_f32_16x16x4_f32(false, a, false, b, (short)0, c, false, false);
}

// C(32x32) = X(32x32) * Y(32x32), all row-major f32 in LDS, one wave (32 lanes)
__device__ __forceinline__ void gemm32_nn(float* C, const float* X, const float* Ym, int lane) {
  const int m8 = lane >> 4;      // half-wave selector
  const int ln = lane & 15;
  for (int r = 0; r < 2; ++r) {
    for (int c = 0; c < 2; ++c) {
      v8f acc;
#pragma unroll
      for (int q = 0; q < 8; ++q) acc[q] = 0.0f;
#pragma unroll
      for (int kk = 0; kk < 8; ++kk) {
        int k0 = kk * 4 + 2 * m8;
        v2f a, b;
        a[0] = X[(r * 16 + ln) * 32 + k0];
        a[1] = X[(r * 16 + ln) * 32 + k0 + 1];
        b[0] = Ym[(k0) * 32 + c * 16 + ln];
        b[1] = Ym[(k0 + 1) * 32 + c * 16 + ln];
        acc = wmma_f32_k4(acc, a, b);
      }
#pragma unroll
      for (int q = 0; q < 8; ++q)
        C[(r * 16 + q + 8 * m8) * 32 + c * 16 + ln] = acc[q];
    }
  }
}

// ---------------------------------------------------------------------------
// Kernel 1: per-Hamiltonian preprocessing.
// H_herm = i*(H-H^T)/2 = i*A, A real antisym. With Y = A/2 (||Y||~0.03):
//   G     = -tanh(Y)                (real antisymmetric)
//   logeta= 0.5 * tr(log cosh(Y)) = 0.5*(trY2/2 - trY4/12 + trY6/45)
// Series via WMMA f32 GEMMs. One wave per Hamiltonian.
// ---------------------------------------------------------------------------
__global__ void gs_precompute_kernel(const double* __restrict__ H1,
                                     const double* __restrict__ H2,
                                     float* __restrict__ Gws,
                                     double* __restrict__ logeta) {
  const int j = blockIdx.x;
  const double* H = j ? H2 : H1;
  __shared__ float Y[1024], Y2[1024], Y4[1024], T[1024], P[1024], TH[1024];
  const int lane = threadIdx.x;  // 32 threads

  __builtin_prefetch(H, 0, 0);

  for (int idx = lane; idx < 1024; idx += 32) {
    int a = idx >> 5, b = idx & 31;
    Y[idx] = (float)((H[a * 32 + b] - H[b * 32 + a]) * 0.25);
  }
  // single wave: LDS ops are in-order, no barrier needed
  gemm32_nn(Y2, Y, Y, lane);
  gemm32_nn(Y4, Y2, Y2, lane);
  for (int idx = lane; idx < 1024; idx += 32) {
    int a = idx >> 5, b = idx & 31;
    T[idx] = ((a == b) ? (-1.0f / 3.0f) : 0.0f) + (2.0f / 15.0f) * Y2[idx] -
             (17.0f / 315.0f) * Y4[idx];
  }
  gemm32_nn(P, Y2, T, lane);
  for (int idx = lane; idx < 1024; idx += 32) {
    int a = idx >> 5, b = idx & 31;
    T[idx] = ((a == b) ? 1.0f : 0.0f) + P[idx];
  }
  gemm32_nn(TH, Y, T, lane);  // TH = tanh(Y)
  for (int idx = lane; idx < 1024; idx += 32) Gws[j * 1024 + idx] = -TH[idx];

  if (lane == 0) {
    double tr2 = 0.0, tr4 = 0.0, tr6 = 0.0;
    for (int a = 0; a < 32; ++a) { tr2 += (double)Y2[a * 33]; tr4 += (double)Y4[a * 33]; }
    for (int idx = 0; idx < 1024; ++idx) {
      int a = idx >> 5, b = idx & 31;
      tr6 += (double)Y2[idx] * (double)Y4[b * 32 + a];
    }
    logeta[j] = 0.5 * (tr2 / 2.0 - tr4 / 12.0 + tr6 / 45.0);
  }
}

// ---------------------------------------------------------------------------
// Kernel 2: per-(sample, Hamiltonian) Pfaffian with blocked Householder.
// Everything LDS-resident; rank-2k panel updates via v_wmma_f32_16x16x4_f32;
// Gh staged global->LDS through the gfx1250 async-to-LDS path (ASYNCcnt).
// grid = (2, BATCH), block = 128 (4 waves).
// ---------------------------------------------------------------------------
__global__ void __launch_bounds__(128)
gs_pfaffian_kernel(const int* __restrict__ xg,
                   const float* __restrict__ Gf,
                   const double* __restrict__ logeta,
                   double* __restrict__ lo) {
  const int jham = blockIdx.x;
  const int s = blockIdx.y;
  const int tid = threadIdx.x;
  const float* Gh = Gf + jham * 1024;

  __shared__ float Ar[NPF * NPF], Ai[NPF * NPF];        // 64x64 complex pfmat
  __shared__ float Vr[NPF * 16], Vi[NPF * 16];          // Householder panel V
  __shared__ float Wr[NPF * 16], Wi[NPF * 16];          // Householder panel W
  __shared__ float Xr[NSPIN * NMAJ], Xi[NSPIN * NMAJ];  // X = M^-1 L^H (16x32)
  __shared__ __align__(16) float GhL[NMAJ * NMAJ];      // staged G (4 KB)
  __shared__ float colr[NPF], coli[NPF];
  __shared__ float s1r[16], s1i[16], s2r[16], s2i[16];
  __shared__ float sg[16];
  __shared__ float fscal[4];   // [0]=tau, [1]=prod(sg)
  __shared__ double scal[4];   // [0]=logpf, [1]=sign.re, [2]=sign.im

  __builtin_amdgcn_s_wait_tensorcnt(0);
  __builtin_prefetch(&xg[s * NSPIN], 0, 0);

  // ---- stage Gh into LDS via async DMA (ASYNCcnt path, no VGPR round-trip) ----
#if __has_builtin(__builtin_amdgcn_global_load_async_to_lds_b128)
  {
    // builtin expects addrspace(1) int4* source, addrspace(3) int4* dest
    __attribute__((address_space(1))) i4v* gsrc =
        (__attribute__((address_space(1))) i4v*)Gh;
    __attribute__((address_space(3))) i4v* ldst =
        (__attribute__((address_space(3))) i4v*)GhL;
    // 4096 B total: 128 threads x 16 B x 2 passes
    __builtin_amdgcn_global_load_async_to_lds_b128(gsrc + tid, ldst + tid, 0, 0);
    __builtin_amdgcn_global_load_async_to_lds_b128(gsrc + 128 + tid, ldst + 128 + tid, 0, 0);
#if __has_builtin(__builtin_amdgcn_s_wait_asynccnt)
    __builtin_amdgcn_s_wait_asynccnt(0);
#else
    asm volatile("s_wait_asynccnt 0x0" ::: "memory");
#endif
  }
#else
  for (int idx = tid; idx < NMAJ * NMAJ; idx += 128) GhL[idx] = Gh[idx];
#endif

  // ---- signs: zz_c = x_c * x_{c+1 mod n}; flip last ----
  if (tid == 0) {
    float prod = 1.0f;
    for (int c = 0; c < NSPIN; ++c) {
      int zz = xg[s * NSPIN + c] * xg[s * NSPIN + ((c + 1) & 15)];
      float sv = (float)zz;
      if (c == 15) sv = -sv;
      sg[c] = sv;
      prod *= sv;
    }
    fscal[1] = prod;
    scal[0] = 0.0; scal[1] = 1.0; scal[2] = 0.0;
  }
  __syncthreads();

  // ---- closed-form cyclic bidiagonal solve: M X = L^H, M = (P - S)/2 ----
  // row c: -(sg_c/2) x_c + (1/2) x_{c+1} = b_c  =>  x_{c+1} = 2 b_c + sg_c x_c
  if (tid < NMAJ) {
    const int j2 = tid;
    const float c2 = 0.70710678118654752f;
    float xr = 0.0f, xi = 0.0f;
    for (int c = 0; c < NSPIN; ++c) {
      float br = 0.0f, bi = 0.0f;
      if (j2 == 2 * c + 1) bi = sg[c] * c2;          // conj(top_c) = i*sg_c/sqrt2
      if (j2 == ((2 * c + 2) & 31)) br = c2;         // conj(bot)   = 1/sqrt2
      float nxr = 2.0f * br + sg[c] * xr;
      float nxi = 2.0f * bi + sg[c] * xi;
      xr = nxr; xi = nxi;
    }
    float denom = 1.0f - fscal[1];
    if (fabsf(denom) < 1e-6f) denom = 1e-6f;
    xr /= denom; xi /= denom;  // x_0 from cyclic closure
    for (int c = 0; c < NSPIN; ++c) {
      Xr[c * NMAJ + j2] = xr; Xi[c * NMAJ + j2] = xi;
      float br = 0.0f, bi = 0.0f;
      if (j2 == 2 * c + 1) bi = sg[c] * c2;
      if (j2 == ((2 * c + 2) & 31)) br = c2;
      float nxr = 2.0f * br + sg[c] * xr;
      float nxi = 2.0f * bi + sg[c] * xi;
      xr = nxr; xi = nxi;
    }
  }
  __syncthreads();

  // ---- build pfmat = [[Gh, -I],[I, D]], D = (Gs - Gs^T)/2 ----
  for (int idx = tid; idx < NPF * NPF; idx += 128) {
    int a = idx >> 6, b = idx & 63;
    float re = 0.0f, im = 0.0f;
    if (a < NMAJ && b < NMAJ) {
      re = GhL[a * NMAJ + b];  // real antisymmetric already
    } else if (a < NMAJ) {
      re = ((b - NMAJ) == a) ? -1.0f : 0.0f;
    } else if (b < NMAJ) {
      re = ((a - NMAJ) == b) ? 1.0f : 0.0f;
    } else {
      const float c2 = 0.70710678118654752f;
      int aa = a - NMAJ, bb = b - NMAJ;
      // Gs[p][q] = delta - 2 * f(p) * X[p/2][q],  f = 1/sqrt2 or i/sqrt2
      int ka = aa >> 1, kb = bb >> 1;
      float x1r = Xr[ka * NMAJ + bb], x1i = Xi[ka * NMAJ + bb];
      float r1, i1;
      if (aa & 1) { r1 = -c2 * x1i; i1 = c2 * x1r; } else { r1 = c2 * x1r; i1 = c2 * x1i; }
      float g1r = ((aa == bb) ? 1.0f : 0.0f) - 2.0f * r1;
      float g1i = -2.0f * i1;
      float x2r = Xr[kb * NMAJ + aa], x2i = Xi[kb * NMAJ + aa];
      float r2, i2;
      if (bb & 1) { r2 = -c2 * x2i; i2 = c2 * x2r; } else { r2 = c2 * x2r; i2 = c2 * x2i; }
      float g2r = ((aa == bb) ? 1.0f : 0.0f) - 2.0f * r2;
      float g2i = -2.0f * i2;
      re = 0.5f * (g1r - g2r);
      im = 0.5f * (g1i - g2i);
    }
    Ar[idx] = re; Ai[idx] = im;
  }
  __syncthreads();

  // ---- blocked complex Householder Pfaffian, 62 steps, NB=16 panels ----
  for (int i = 0; i < NPF - 2; ++i) {
    const int p = i & ~15;
    const int ip = i - p;

    if (ip == 0) {  // fresh panel
      for (int idx = tid; idx < NPF * 16; idx += 128) {
        Vr[idx] = 0.0f; Vi[idx] = 0.0f; Wr[idx] = 0.0f; Wi[idx] = 0.0f;
      }
    }
    __syncthreads();

    // corrected column i:  a = A[:,i] + V*W^T[:,i] - W*V^T[:,i]
    if (tid < NPF) {
      int t = tid;
      float cr = Ar[t * NPF + i], ci = Ai[t * NPF + i];
      for (int q = 0; q < ip; ++q) {
        float vr_ = Vr[t * 16 + q], vi_ = Vi[t * 16 + q];
        float wr_ = Wr[t * 16 + q], wi_ = Wi[t * 16 + q];
        float wir = Wr[i * 16 + q], wii = Wi[i * 16 + q];
        float vir = Vr[i * 16 + q], vii = Vi[i * 16 + q];
        cr += vr_ * wir - vi_ * wii - (wr_ * vir - wi_ * vii);
        ci += vr_ * wii + vi_ * wir - (wr_ * vii + wi_ * vir);
      }
      colr[t] = cr; coli[t] = ci;
    }
    __syncthreads();

    // Householder of col at pivot k=i+1 (thread 0; n=64 is tiny)
    if (tid == 0) {
      const int k = i + 1;
      float sigma = 0.0f;
      for (int t = k + 1; t < NPF; ++t) sigma += colr[t] * colr[t] + coli[t] * coli[t];
      float xnr = colr[k], xni = coli[k];
      float xa = sqrtf(xnr * xnr + xni * xni);
      float nx = sqrtf(xa * xa + sigma);
      float phr = 1.0f, phi = 0.0f;
      if (xa > 0.0f) { phr = xnr / xa; phi = xni / xa; }
      float vnr = xnr + phr * nx, vni = xni + phi * nx;
      float alr = -phr * nx, ali = -phi * nx;
      float tau;
      if (sigma == 0.0f) {
        tau = 0.0f; alr = xnr; ali = xni;
        for (int t = 0; t < NPF; ++t) { colr[t] = 0.0f; coli[t] = 0.0f; }
      } else {
        tau = 2.0f;
        float inv = 1.0f / sqrtf(vnr * vnr + vni * vni + sigma);
        for (int t = 0; t < NPF; ++t) {
          float vr_ = (t == k) ? vnr : ((t > k) ? colr[t] : 0.0f);
          float vi_ = (t == k) ? vni : ((t > k) ? coli[t] : 0.0f);
          colr[t] = vr_ * inv; coli[t] = vi_ * inv;
        }
      }
      fscal[0] = tau;
      // sign/log accumulation (f64): log|1-tau|=0; sign *= (1-tau)/|1-tau|
      double str = (tau == 0.0f) ? 1.0 : -1.0;
      double nsr = scal[1] * str, nsi = scal[2] * str;
      double lr = scal[0];
      if ((i & 1) == 0) {
        double aa = sqrt((double)alr * alr + (double)ali * ali);
        if (aa < 1e-300) aa = 1e-300;
        lr += log(aa);
        double asr = -(double)alr / aa, asi = -(double)ali / aa;
        double t2r = nsr * asr - nsi * asi, t2i = nsr * asi + nsi * asr;
        nsr = t2r; nsi = t2i;
      }
      scal[0] = lr; scal[1] = nsr; scal[2] = nsi;
    }
    __syncthreads();

    // stash v; panel inner products s1 = W^T conj(v), s2 = V^T conj(v)
    if (tid < NPF) { Vr[tid * 16 + ip] = colr[tid]; Vi[tid * 16 + ip] = coli[tid]; }
    if (tid >= 64 && tid < 80) {
      int q = tid - 64;
      float ar = 0.0f, ai = 0.0f, br = 0.0f, bi = 0.0f;
      if (q < ip) {
        for (int j2 = 0; j2 < NPF; ++j2) {
          float vr_ = colr[j2], vi_ = -coli[j2];
          ar += Wr[j2 * 16 + q] * vr_ - Wi[j2 * 16 + q] * vi_;
          ai += Wr[j2 * 16 + q] * vi_ + Wi[j2 * 16 + q] * vr_;
          br += Vr[j2 * 16 + q] * vr_ - Vi[j2 * 16 + q] * vi_;
          bi += Vr[j2 * 16 + q] * vi_ + Vi[j2 * 16 + q] * vr_;
        }
      }
      s1r[q] = ar; s1i[q] = ai; s2r[q] = br; s2i[q] = bi;
    }
    __syncthreads();

    // w = tau * (A conj(v) + V s1 - W s2)   (cooperative matvec, 64 rows)
    if (tid < NPF) {
      int t = tid;
      float ar = 0.0f, ai = 0.0f;
      for (int j2 = 0; j2 < NPF; ++j2) {
        float vr_ = colr[j2], vi_ = -coli[j2];
        ar += Ar[t * NPF + j2] * vr_ - Ai[t * NPF + j2] * vi_;
        ai += Ar[t * NPF + j2] * vi_ + Ai[t * NPF + j2] * vr_;
      }
      for (int q = 0; q < ip; ++q) {
        ar += Vr[t * 16 + q] * s1r[q] - Vi[t * 16 + q] * s1i[q] -
              (Wr[t * 16 + q] * s2r[q] - Wi[t * 16 + q] * s2i[q]);
        ai += Vr[t * 16 + q] * s1i[q] + Vi[t * 16 + q] * s1r[q] -
              (Wr[t * 16 + q] * s2i[q] + Wi[t * 16 + q] * s2r[q]);
      }
      float tau = fscal[0];
      Wr[t * 16 + ip] = tau * ar; Wi[t * 16 + ip] = tau * ai;
    }
    __syncthreads();

    // ---- panel boundary: A += V W^T - W V^T via WMMA (16x16 tiles, K=16) ----
    if (ip == 15 || i == NPF - 3) {
      const int wv = tid >> 5;
      const int lane = tid & 31;
      const int m8 = lane >> 4;
      const int ln = lane & 15;
      for (int t = wv; t < 16; t += 4) {  // wave-uniform tile loop
        int r = t >> 2, c = t & 3;
        v8f cr, ci;
#pragma unroll
        for (int q = 0; q < 8; ++q) {
          int row = r * 16 + q + 8 * m8;
          cr[q] = Ar[row * NPF + c * 16 + ln];
          ci[q] = Ai[row * NPF + c * 16 + ln];
        }
#pragma unroll
        for (int kk = 0; kk < 4; ++kk) {
          int k0 = kk * 4 + 2 * m8;
          int arow = (r * 16 + ln) * 16;
          int brow = (c * 16 + ln) * 16;
          v2f avr, avi, awr, awi, bvr, bvi, bwr, bwi;
          avr[0] = Vr[arow + k0]; avr[1] = Vr[arow + k0 + 1];
          avi[0] = Vi[arow + k0]; avi[1] = Vi[arow + k0 + 1];
          awr[0] = Wr[arow + k0]; awr[1] = Wr[arow + k0 + 1];
          awi[0] = Wi[arow + k0]; awi[1] = Wi[arow + k0 + 1];
          bvr[0] = Vr[brow + k0]; bvr[1] = Vr[brow + k0 + 1];
          bvi[0] = Vi[brow + k0]; bvi[1] = Vi[brow + k0 + 1];
          bwr[0] = Wr[brow + k0]; bwr[1] = Wr[brow + k0 + 1];
          bwi[0] = Wi[brow + k0]; bwi[1] = Wi[brow + k0 + 1];
          // real: Vr Wr^T - Vi Wi^T - Wr Vr^T + Wi Vi^T
          cr = wmma_f32_k4(cr, avr, bwr);
          cr = wmma_f32_k4(cr, avi, -bwi);
          cr = wmma_f32_k4(cr, awr, -bvr);
          cr = wmma_f32_k4(cr, awi, bvi);
          // imag: Vr Wi^T + Vi Wr^T - Wr Vi^T - Wi Vr^T
          ci = wmma_f32_k4(ci, avr, bwi);
          ci = wmma_f32_k4(ci, avi, bwr);
          ci = wmma_f32_k4(ci, awr, -bvi);
          ci = wmma_f32_k4(ci, awi, -bvr);
        }
#pragma unroll
        for (int q = 0; q < 8; ++q) {
          int row = r * 16 + q + 8 * m8;
          Ar[row * NPF + c * 16 + ln] = cr[q];
          Ai[row * NPF + c * 16 + ln] = ci[q];
        }
      }
      __syncthreads();
    }
  }

  // ---- finalize: pf term A[62,63]; lo = logeta + log(sign) + logabs ----
  if (tid == 0) {
    float ar = Ar[62 * NPF + 63], ai = Ai[62 * NPF + 63];
    double aa = sqrt((double)ar * ar + (double)ai * ai);
    if (aa < 1e-300) aa = 1e-300;
    double lp = scal[0] + log(aa);
    double crn = (double)ar / aa, cin = (double)ai / aa;
    double nr = scal[1] * crn - scal[2] * cin;
    double ni = scal[1] * cin + scal[2] * crn;
    // sign_pref = (-1)^(64/2) = +1
    double lor = logeta[jham] + 0.5 * log(nr * nr + ni * ni) + lp;
    double loi = atan2(ni, nr);
    lo[(s * 2 + jham) * 2 + 0] = lor;
    lo[(s * 2 + jham) * 2 + 1] = loi;
  }
}

// ---------------------------------------------------------------------------
// Kernel 3: complex logsumexp over the two branches (with s_prod phase).
// ---------------------------------------------------------------------------
__global__ void gs_combine_kernel(const int* __restrict__ xg,
                                  const double* __restrict__ lo,
                                  double* __restrict__ out) {
  int s = blockIdx.x * blockDim.x + threadIdx.x;
  if (s >= BATCH) return;
  const double PI = 3.14159265358979323846;
  double a0r = lo[(s * 2 + 0) * 2], a0i = lo[(s * 2 + 0) * 2 + 1];
  double a1r = lo[(s * 2 + 1) * 2], a1i = lo[(s * 2 + 1) * 2 + 1];
  int pr = 1;
  for (int c = 0; c < NSPIN; ++c) pr *= xg[s * NSPIN + c];
  if (pr < 0) a1i += PI;  // log(-1) = i*pi
  double m = fmax(a0r, a1r);
  double e0 = exp(a0r - m), e1 = exp(a1r - m);
  double sr = e0 * cos(a0i) + e1 * cos(a1i);
  double si = e0 * sin(a0i) + e1 * sin(a1i);
  out[2 * s + 0] = 0.5 * log(sr * sr + si * si) + m;
  out[2 * s + 1] = atan2(si, sr);
}

extern "C" void kernel_launch(void* const* d_in, const int* in_sizes, int n_in,
                              void* d_out, int out_size, void* d_ws, size_t ws_size,
                              hipStream_t stream) {
  const int*    x  = (const int*)d_in[0];
  const double* H1 = (const double*)d_in[1];
  const double* H2 = (const double*)d_in[2];

  // workspace layout
  float*  Gf     = (float*)d_ws;                          // 2 * 32*32 f32 = 8 KB
  double* logeta = (double*)((char*)d_ws + 8192);         // 2 doubles
  double* lo     = (double*)((char*)d_ws + 8192 + 64);    // 2*B complex f64 = 8 KB
  double* outd   = (double*)d_out;                        // B complex128

  gs_precompute_kernel<<<2, 32, 0, stream>>>(H1, H2, Gf, logeta);
  gs_pfaffian_kernel<<<dim3(2, BATCH), 128, 0, stream>>>(x, Gf, logeta, lo);
  gs_combine_kernel<<<1, 256, 0, stream>>>(x, lo, outd);
}